// Oscillator_42047729828044
// MI455X (gfx1250) — compile-verified
//
#include <hip/hip_runtime.h>
#include <math.h>

typedef __attribute__((ext_vector_type(2))) float v2f;
typedef __attribute__((ext_vector_type(8))) float v8f;

namespace {
constexpr int kB   = 64;
constexpr int kF   = 4096;
constexpr int kHop = 64;
constexpr float  kInvSR  = 1.0f / 16000.0f;
constexpr double kInvSRd = 1.0  / 16000.0;
constexpr float  kInv2Pi = 0.15915494309189535f;  // 1/(2*pi), inline const on gfx1250
}

// ---------------------------------------------------------------------------
// Kernel 1: per-row exclusive prefix scan of frequency (F=4096), producing
//   step[b,f] = freq[b,f] / SR                       (cycles per sample)
//   base[b,f] = fract( (HOP * S_f - freq[b,0]) / SR) (cycles, reduced mod 1)
// One 256-thread workgroup per batch row; scan partials in LDS (f64 accum).
// ---------------------------------------------------------------------------
__global__ __launch_bounds__(256) void osc_scan_kernel(
    const float* __restrict__ freq, float* __restrict__ base_out,
    float* __restrict__ step_out) {
  const int b = blockIdx.x;
  const int i = threadIdx.x;
  const float* row = freq + (size_t)b * kF;

  float vals[16];
  const float4* r4 = reinterpret_cast<const float4*>(row) + i * 4;
#pragma unroll
  for (int q = 0; q < 4; ++q) {
    float4 t = r4[q];
    vals[4 * q + 0] = t.x; vals[4 * q + 1] = t.y;
    vals[4 * q + 2] = t.z; vals[4 * q + 3] = t.w;
  }
  double lsum = 0.0;
#pragma unroll
  for (int j = 0; j < 16; ++j) lsum += (double)vals[j];

  __shared__ double s[256];
  s[i] = lsum;
  for (int off = 1; off < 256; off <<= 1) {   // Hillis-Steele inclusive scan
    __syncthreads();
    double t = (i >= off) ? s[i - off] : 0.0;
    __syncthreads();
    s[i] += t;
  }
  __syncthreads();

  double run = (i == 0) ? 0.0 : s[i - 1];     // exclusive prefix for this chunk
  const double f0 = (double)row[0];
#pragma unroll
  for (int j = 0; j < 16; ++j) {
    const int f = i * 16 + j;
    double bc = (run * (double)kHop - f0) * kInvSRd;
    bc -= floor(bc);                          // reduce mod 1 cycle
    base_out[(size_t)b * kF + f] = (float)bc;
    step_out[(size_t)b * kF + f] = vals[j] * kInvSR;
    run += (double)vals[j];
  }
}

// ---------------------------------------------------------------------------
// Kernel 2: streaming pass. Each wave owns 16 segments x 64 samples (1024
// elems). Phase tiles built with V_WMMA_F32_16X16X4_F32 as a rank-2 product:
//   D[m,n] = step_m * (toff+n+1) + base_m * 1
// A[16x4] layout (32-bit A): lanes 0-15 hold K=0 (v.x) and K=1 (v.y);
// lanes 16-31 hold K=2,3 (zeros). B[4x16]: VGPR0 lanes 0-15 = row K=0,
// lanes 16-31 = row K=1; VGPR1 = rows K=2,3 (zeros).
// D layout: VGPR r, lanes 0-15 -> M=r, lanes 16-31 -> M=r+8, N = lane%16.
// Then out = v_sin_f32(phase_cycles + pm * (1/2pi))  [v_sin is sin(2pi*x)].
// ---------------------------------------------------------------------------
__global__ __launch_bounds__(256) void osc_main_kernel(
    const float* __restrict__ pm, const float* __restrict__ base_in,
    const float* __restrict__ step_in, float* __restrict__ out) {
  const int blk   = blockIdx.x;
  const int b     = blk >> 5;          // 32 blocks per batch row
  const int sgblk = blk & 31;
  const int wave  = threadIdx.x >> 5;  // 8 waves per block
  const int lane  = threadIdx.x & 31;
  const int f0    = sgblk * 128 + wave * 16;
  const int n     = lane & 15;
  const bool lo   = lane < 16;

  const int rowF = b * kF;
  const float stepv = step_in[rowF + f0 + n];
  const float basev = base_in[rowF + f0 + n];
  v2f a;
  a.x = lo ? stepv : 0.0f;   // A[m,0] = step_m   (lanes 0-15)
  a.y = lo ? basev : 0.0f;   // A[m,1] = base_m

  const int fl = f0 + (lo ? 0 : 8);                 // M offset per half-wave
  const unsigned idx0 = (unsigned)(rowF + fl) * kHop + (unsigned)n;

#pragma unroll
  for (int toff = 0; toff < kHop; toff += 16) {
    v2f bb;
    bb.x = lo ? (float)(toff + n + 1) : 1.0f;       // B[0,n]=t+1 ; B[1,n]=1
    bb.y = 0.0f;                                    // B[2,*]=B[3,*]=0
    v8f c = {};
    c = __builtin_amdgcn_wmma_f32_16x16x4_f32(
        /*neg_a=*/false, a, /*neg_b=*/false, bb,
        /*c_mod=*/(short)0, c, /*reuse_a=*/false, /*reuse_b=*/false);
#pragma unroll
    for (int r = 0; r < 8; ++r) {
      const unsigned idx = idx0 + (unsigned)(r * kHop + toff);
      const float ph = c[r] + pm[idx] * kInv2Pi;    // cycles
      out[idx] = __builtin_amdgcn_sinf(ph);         // v_sin_f32: sin(2pi*x)
    }
  }
}

extern "C" void kernel_launch(void* const* d_in, const int* in_sizes, int n_in,
                              void* d_out, int out_size, void* d_ws, size_t ws_size,
                              hipStream_t stream) {
  (void)in_sizes; (void)n_in; (void)out_size; (void)ws_size;
  const float* freq = (const float*)d_in[0];   // [64, 4096]
  const float* pm   = (const float*)d_in[1];   // [64, 262144]
  float* out = (float*)d_out;                  // [64, 262144]

  float* base_ws = (float*)d_ws;               // [64*4096]
  float* step_ws = base_ws + kB * kF;          // [64*4096]  (2 MB total)

  osc_scan_kernel<<<kB, 256, 0, stream>>>(freq, base_ws, step_ws);
  // 64 rows * 32 segment-blocks = 2048 blocks; 8 waves each, wave = 16 segs.
  osc_main_kernel<<<kB * 32, 256, 0, stream>>>(pm, base_ws, step_ws, out);
}